// GroupingUnit_1039382085915
// MI455X (gfx1250) — compile-verified
//
#include <hip/hip_runtime.h>
#include <hip/hip_bf16.h>

// Problem constants (match reference)
#define Bn 16
#define Cc 512
#define Nn 4096   // H*W = 64*64
#define Pp 32

typedef __attribute__((ext_vector_type(2))) float v2f;
typedef __attribute__((ext_vector_type(8))) float v8f;

// ---------------------------------------------------------------------------
// gfx1250 async Memory->LDS helpers (VGLOBAL GLOBAL_LOAD_ASYNC_TO_LDS_*,
// tracked by ASYNCcnt). LDS operand is the 32-bit LDS byte offset = low 32
// bits of the generic shared-aperture pointer.
// ---------------------------------------------------------------------------
__device__ __forceinline__ void async_ld_b128(const void* g, void* l) {
    asm volatile("global_load_async_to_lds_b128 %0, %1, off"
                 :: "v"((unsigned)(size_t)l),
                    "v"((unsigned long long)(size_t)g)
                 : "memory");
}
__device__ __forceinline__ void async_ld_b32(const void* g, void* l) {
    asm volatile("global_load_async_to_lds_b32 %0, %1, off"
                 :: "v"((unsigned)(size_t)l),
                    "v"((unsigned long long)(size_t)g)
                 : "memory");
}
__device__ __forceinline__ void wait_async_0() {
#if __has_builtin(__builtin_amdgcn_s_wait_asynccnt)
    __builtin_amdgcn_s_wait_asynccnt(0);
#else
    asm volatile("s_wait_asynccnt 0x0" ::: "memory");
#endif
}
__device__ __forceinline__ void wait_async_10() {
#if __has_builtin(__builtin_amdgcn_s_wait_asynccnt)
    __builtin_amdgcn_s_wait_asynccnt(10);
#else
    asm volatile("s_wait_asynccnt 0xa" ::: "memory");
#endif
}

// ---------------------------------------------------------------------------
// Workspace layout (floats):
//   [0,32)    c_sq[p]      = sum_c centers[p][c]^2
//   [32,64)   inv_beta[p]  = 1/sigmoid(smooth_factor[p])
//   [64,96)   inv_sigma[p] = sqrt(2/beta[p])
//   [96,608)  sum_ass[b*P+p] (atomically accumulated by k_assign)
// ---------------------------------------------------------------------------

__global__ __launch_bounds__(64) void k_pre(const float* __restrict__ weight,
                                            const float* __restrict__ sf,
                                            float* __restrict__ ws) {
    int t = threadIdx.x;
    if (t < Pp) {
        float s = 0.f;
        for (int c = 0; c < Cc; ++c) {
            float w = weight[t * Cc + c];
            s += w * w;
        }
        ws[t] = s;                               // c_sq
        float beta = 1.f / (1.f + __expf(-sf[t]));
        ws[32 + t] = 1.f / beta;                 // inv_beta
        ws[64 + t] = sqrtf(2.f / beta);          // inv_sigma
    }
    for (int i = t; i < Bn * Pp; i += 64) ws[96 + i] = 0.f;  // sum_ass = 0
}

// ---------------------------------------------------------------------------
// Kernel 1: cx GEMM (centers x X) + x^2 + softmax over P -> assign_map,
//           plus atomic accumulation of sum_ass[b][p].
// grid = (Nn/128, Bn), block = 256 (8 waves); each wave owns 16 columns and
// computes the full 32(P) x 16(n) logit tile via two f32 WMMA accumulators.
// Centers are async-transposed into LDS pair-interleaved [(k/2)][p][2] so an
// A-fragment is one aligned ds_load_b64 (conflict-free across the wave).
// ---------------------------------------------------------------------------
__global__ __launch_bounds__(256) void k_assign(const float* __restrict__ x,
                                                const float* __restrict__ weight,
                                                const float* __restrict__ ws,
                                                float* __restrict__ assign_out,
                                                float* __restrict__ sum_ass) {
    __shared__ float CsP[Cc * Pp];   // [(k/2)][p][2] : exactly 64 KB
    const int b   = blockIdx.y;
    const int tid = threadIdx.x;

    // Async transposing fill: coalesced global reads, per-lane LDS scatter.
    for (int idx = tid; idx < Cc * Pp; idx += 256) {
        int p = idx >> 9;            // idx / 512
        int k = idx & (Cc - 1);      // idx % 512
        async_ld_b32(weight + idx, &CsP[(((k >> 1) * Pp) + p) * 2 + (k & 1)]);
    }
    wait_async_0();
    __syncthreads();

    const int wave = tid >> 5;
    const int lane = tid & 31;
    const int l16  = lane & 15;
    const int half = lane >> 4;
    const int koff = half * 2;
    const int n0   = blockIdx.x * 128 + wave * 16;

    const float* xb   = x + (size_t)b * Cc * Nn + n0 + l16;
    const v2f*   CsPv = (const v2f*)CsP;   // one v2f = (K, K+1) for a given p

    v8f acc0 = {};   // p = 0..15
    v8f acc1 = {};   // p = 16..31
    float sq = 0.f;  // partial sum_k x[k][n]^2 (this lane covers k = 4t+koff,+1)

    for (int k = 0; k < Cc; k += 4) {
        // A fragments (16x4 f32 layout: v0 = K=k+koff, v1 = K=k+koff+1)
        int pair = ((k + koff) >> 1) * Pp;
        v2f a0 = CsPv[pair + l16];        // rows p = 0..15
        v2f a1 = CsPv[pair + 16 + l16];   // rows p = 16..31
        // B fragment from global (row k+koff / k+koff+1, 16 consecutive cols)
        float b0 = xb[(size_t)(k + koff) * Nn];
        float b1 = xb[(size_t)(k + koff + 1) * Nn];
        v2f bf; bf.x = b0; bf.y = b1;
        sq += b0 * b0 + b1 * b1;

        acc0 = __builtin_amdgcn_wmma_f32_16x16x4_f32(false, a0, false, bf,
                                                     (short)0, acc0, false, false);
        acc1 = __builtin_amdgcn_wmma_f32_16x16x4_f32(false, a1, false, bf,
                                                     (short)0, acc1, false, false);
    }

    // Full x^2 per column: lane L has k≡{0,1} mod 4, lane L+16 has k≡{2,3}.
    sq += __shfl_xor(sq, 16);

    const float* c_sq     = ws;
    const float* inv_beta = ws + 32;

    float lg[16];
    float mx = -1e30f;
#pragma unroll
    for (int i = 0; i < 8; ++i) {
        int p0 = half * 8 + i;       // acc0: lanes<16 -> p=i, lanes>=16 -> p=8+i
        int p1 = p0 + 16;            // acc1
        float v0 = fminf(2.f * acc0[i] - sq - c_sq[p0], 0.f) * inv_beta[p0];
        float v1 = fminf(2.f * acc1[i] - sq - c_sq[p1], 0.f) * inv_beta[p1];
        lg[i]     = v0;
        lg[8 + i] = v1;
        mx = fmaxf(mx, fmaxf(v0, v1));
    }
    mx = fmaxf(mx, __shfl_xor(mx, 16));   // other 16 p's live in cross-half lane

    float ssum = 0.f;
#pragma unroll
    for (int i = 0; i < 16; ++i) {
        lg[i] = __expf(lg[i] - mx);
        ssum += lg[i];
    }
    ssum += __shfl_xor(ssum, 16);
    float inv = 1.f / ssum;

    float* aout = assign_out + (size_t)b * Pp * Nn + n0 + l16;
#pragma unroll
    for (int i = 0; i < 16; ++i) {
        int p = (i < 8) ? (half * 8 + i) : (16 + half * 8 + (i - 8));
        float a = lg[i] * inv;
        aout[(size_t)p * Nn] = a;
        float r = a;
#pragma unroll
        for (int off = 1; off < 16; off <<= 1) r += __shfl_xor(r, off);
        if (l16 == 0) atomicAdd(&sum_ass[b * Pp + p], r);
    }
}

// ---------------------------------------------------------------------------
// Kernel 2: qx = assign x X^T per batch (M=P=32, K=N=4096, Ncols=C).
// Double-buffered async-to-LDS staging (10 B128 transfers per thread per
// 64-wide K chunk) overlapped with the WMMA K-loop of the previous chunk.
// Rows padded to 68 floats: 16B-aligned rows for B128 LDS writes and 8B-
// aligned v2f fragment reads. Finalize (qx/clip(sum)-center)*inv_sigma and
// write TRANSPOSED [b][c][p] (unnormalized) into d_out.
// grid = (C/128, Bn), block = 256 (8 waves x 16 c-columns).
// ---------------------------------------------------------------------------
#define KC    64          // K-chunk (n) width
#define LPAD  68          // padded row length (floats)
#define NCH   (Nn / KC)   // 64 chunks

__global__ __launch_bounds__(256) void k_pool(const float* __restrict__ x,
                                              const float* __restrict__ assign,
                                              const float* __restrict__ weight,
                                              const float* __restrict__ ws,
                                              float* __restrict__ outp) {
    __shared__ float As[2][Pp][LPAD];     // assign tile  [p][n']
    __shared__ float Xs[2][128][LPAD];    // x tile       [c'][n']

    const int b    = blockIdx.y;
    const int c0   = blockIdx.x * 128;
    const int tid  = threadIdx.x;
    const int wave = tid >> 5;
    const int lane = tid & 31;
    const int l16  = lane & 15;
    const int half = lane >> 4;
    const int koff = half * 2;

    const float* ab = assign + (size_t)b * Pp * Nn;
    const float* xb = x + (size_t)b * Cc * Nn + (size_t)c0 * Nn;

    // Issue one chunk's async transfers: As = 32x64, Xs = 128x64 (B128 each).
    auto prefetch = [&](int buf, int nb) {
        for (int t = tid; t < Pp * (KC / 4); t += 256) {        // 2 per thread
            int r = t >> 4, q = (t & 15) * 4;
            async_ld_b128(ab + (size_t)r * Nn + nb + q, &As[buf][r][q]);
        }
        for (int t = tid; t < 128 * (KC / 4); t += 256) {       // 8 per thread
            int r = t >> 4, q = (t & 15) * 4;
            async_ld_b128(xb + (size_t)r * Nn + nb + q, &Xs[buf][r][q]);
        }
    };

    v8f acc0 = {};   // p = 0..15
    v8f acc1 = {};   // p = 16..31

    prefetch(0, 0);
    for (int i = 0; i < NCH; ++i) {
        const int buf = i & 1;
        if (i + 1 < NCH) {
            prefetch(buf ^ 1, (i + 1) * KC);
            wait_async_10();   // async loads complete in order: chunk i is in
        } else {
            wait_async_0();
        }
        __syncthreads();

        const int cl = wave * 16 + l16;
#pragma unroll
        for (int kk = 0; kk < KC; kk += 4) {
            v2f a0 = *(const v2f*)&As[buf][l16][kk + koff];
            v2f a1 = *(const v2f*)&As[buf][16 + l16][kk + koff];
            v2f bf = *(const v2f*)&Xs[buf][cl][kk + koff];
            acc0 = __builtin_amdgcn_wmma_f32_16x16x4_f32(false, a0, false, bf,
                                                         (short)0, acc0, false, false);
            acc1 = __builtin_amdgcn_wmma_f32_16x16x4_f32(false, a1, false, bf,
                                                         (short)0, acc1, false, false);
        }
        __syncthreads();   // protect buf from the next prefetch overwrite
    }

    const float* inv_sigma = ws + 64;
    const float* sum_ass   = ws + 96;
    const int c = c0 + wave * 16 + l16;
    float* ob = outp + (size_t)b * Cc * Pp + (size_t)c * Pp;
#pragma unroll
    for (int i = 0; i < 8; ++i) {
        int p0 = half * 8 + i;
        int p1 = p0 + 16;
        float s0 = fmaxf(sum_ass[b * Pp + p0], 1e-5f);
        float s1 = fmaxf(sum_ass[b * Pp + p1], 1e-5f);
        float v0 = (acc0[i] / s0 - weight[p0 * Cc + c]) * inv_sigma[p0];
        float v1 = (acc1[i] / s1 - weight[p1 * Cc + c]) * inv_sigma[p1];
        ob[p0] = v0;
        ob[p1] = v1;
    }
}

// ---------------------------------------------------------------------------
// Kernel 3: per-(b,p) L2-normalize over C, in place on the [b][c][p] layout.
// One block per batch; p = tid%32 is invariant under the stride-256 loop.
// ---------------------------------------------------------------------------
__global__ __launch_bounds__(256) void k_norm(float* __restrict__ outp) {
    __shared__ float norms[Pp];
    const int b   = blockIdx.x;
    const int tid = threadIdx.x;
    if (tid < Pp) norms[tid] = 0.f;
    __syncthreads();

    float* base = outp + (size_t)b * Cc * Pp;
    float local = 0.f;
    for (int idx = tid; idx < Cc * Pp; idx += 256) {
        float v = base[idx];
        local += v * v;
    }
    atomicAdd(&norms[tid & 31], local);
    __syncthreads();

    float scale = 1.f / fmaxf(sqrtf(norms[tid & 31]), 1e-12f);
    for (int idx = tid; idx < Cc * Pp; idx += 256) base[idx] *= scale;
}

// ---------------------------------------------------------------------------
extern "C" void kernel_launch(void* const* d_in, const int* in_sizes, int n_in,
                              void* d_out, int out_size, void* d_ws, size_t ws_size,
                              hipStream_t stream) {
    const float* x      = (const float*)d_in[0];   // [B,C,H,W] = [B,C,N]
    const float* weight = (const float*)d_in[1];   // [P,C,1,1] = [P,C]
    const float* sf     = (const float*)d_in[2];   // [P]

    float* out        = (float*)d_out;
    float* outputs_t  = out;                         // [B,C,P]
    float* assign_map = out + (size_t)Bn * Cc * Pp;  // [B,P,N]
    float* ws         = (float*)d_ws;
    float* sum_ass    = ws + 96;

    k_pre<<<1, 64, 0, stream>>>(weight, sf, ws);

    dim3 g1(Nn / 128, Bn);
    k_assign<<<g1, 256, 0, stream>>>(x, weight, ws, assign_map, sum_ass);

    dim3 g2(Cc / 128, Bn);
    k_pool<<<g2, 256, 0, stream>>>(x, assign_map, weight, ws, outputs_t);

    k_norm<<<Bn, 256, 0, stream>>>(outputs_t);
}